// GraphSAGELayer_25915832664412
// MI455X (gfx1250) — compile-verified
//
#include <hip/hip_runtime.h>

#define N_NODES   100000
#define IN_FEATS  256
#define OUT_FEATS 256
#define N_EDGES   3200000

typedef __attribute__((ext_vector_type(2))) float v2f;
typedef __attribute__((ext_vector_type(8))) float v8f;

// ---------------------------------------------------------------------------
// h = x @ W + bias   via V_WMMA_F32_16X16X4_F32
// One wave computes one 16x16 tile of h. Block = 512 threads = 16 waves,
// covering all 16 N-tiles of a 16-row stripe (so A loads are shared via cache).
//
// VGPR layouts (CDNA5 ISA 7.12.2):
//   A (16x4, MxK, 2 VGPRs): lanes 0-15 hold M=lane, K={k,k+1};
//                           lanes 16-31 hold M=lane-16, K={k+2,k+3}
//   B (4x16, KxN, 2 VGPRs): lanes 0-15 hold N=lane, K={k,k+1};
//                           lanes 16-31 hold N=lane-16, K={k+2,k+3}
//   C/D (16x16, 8 VGPRs):   VGPR i = row M=i (lanes 0-15), M=i+8 (lanes 16-31)
// ---------------------------------------------------------------------------
__global__ __launch_bounds__(512)
void sage_gemm_wmma(const float* __restrict__ x, const float* __restrict__ W,
                    const float* __restrict__ bias, float* __restrict__ h) {
  const int lane = threadIdx.x & 31;
  const int wave = threadIdx.x >> 5;   // 0..15 -> which N tile
  const int m0   = blockIdx.x * 16;
  const int n0   = wave * 16;
  const int half = lane >> 4;          // 0 = lanes 0-15, 1 = lanes 16-31
  const int l15  = lane & 15;

  v8f c = {};

  // Per-lane base pointers
  const float* xrow = x + (size_t)(m0 + l15) * IN_FEATS + half * 2;   // A frag: float2 at K=k(+2)
  const float* wcol = W + (size_t)(half * 2) * OUT_FEATS + n0 + l15;  // B frag base

#pragma unroll 4
  for (int k = 0; k < IN_FEATS; k += 4) {
    v2f a = *(const v2f*)(xrow + k);                   // global_load_b64, contiguous
    v2f b;
    b.x = wcol[(size_t)k * OUT_FEATS];                 // W[k+2*half   ][n]
    b.y = wcol[(size_t)(k + 1) * OUT_FEATS];           // W[k+2*half+1 ][n]
    c = __builtin_amdgcn_wmma_f32_16x16x4_f32(
        /*neg_a=*/false, a, /*neg_b=*/false, b,
        /*c_mod=*/(short)0, c, /*reuse_a=*/false, /*reuse_b=*/false);
  }

  const float bv = bias[n0 + l15];
  float* hout = h + n0 + l15;
#pragma unroll
  for (int i = 0; i < 8; ++i) {
    const int m = m0 + i + half * 8;
    hout[(size_t)m * OUT_FEATS] = c[i] + bv;
  }
}

// ---------------------------------------------------------------------------
// Zero helper (d_out / deg are poisoned by the harness)
// ---------------------------------------------------------------------------
__global__ void sage_zero_f4(float4* __restrict__ p, int n4) {
  int i = blockIdx.x * blockDim.x + threadIdx.x;
  if (i < n4) p[i] = make_float4(0.f, 0.f, 0.f, 0.f);
}

// ---------------------------------------------------------------------------
// Degree: deg[row] += 1 per edge (f32 atomics, resolve in L2)
// ---------------------------------------------------------------------------
__global__ void sage_degree(const long long* __restrict__ ei, float* __restrict__ deg) {
  int e = blockIdx.x * blockDim.x + threadIdx.x;
  if (e < N_EDGES) atomicAdd(&deg[ei[e]], 1.0f);
}

// ---------------------------------------------------------------------------
// Scatter: out[row] += h[col].  64 threads per edge, float4 gather
// (global_load_b128) + 4x global_atomic_add_f32.  out (102 MB) fits in the
// 192 MB L2, so atomics stay on-chip.
// ---------------------------------------------------------------------------
__global__ __launch_bounds__(256)
void sage_scatter(const float* __restrict__ h, const long long* __restrict__ ei,
                  float* __restrict__ out) {
  const long long tid = (long long)blockIdx.x * blockDim.x + threadIdx.x;
  const long long e   = tid >> 6;        // 64 threads per edge
  const int chunk     = (int)(tid & 63); // which float4 of the 256-wide row
  if (e >= N_EDGES) return;
  const long long row = ei[e];
  const long long col = ei[N_EDGES + e];
  const float4 v = ((const float4*)(h + col * OUT_FEATS))[chunk];
  float* o = out + row * OUT_FEATS + (size_t)chunk * 4;
  atomicAdd(o + 0, v.x);
  atomicAdd(o + 1, v.y);
  atomicAdd(o + 2, v.z);
  atomicAdd(o + 3, v.w);
}

// ---------------------------------------------------------------------------
// Normalize: out[n] *= (deg[n] > 0 ? 1/deg[n] : 0), float4-vectorized
// ---------------------------------------------------------------------------
__global__ void sage_scale(float* __restrict__ out, const float* __restrict__ deg) {
  int i = blockIdx.x * blockDim.x + threadIdx.x;  // over N_NODES*64 float4s
  if (i >= N_NODES * (OUT_FEATS / 4)) return;
  const float d = deg[i >> 6];
  const float s = d > 0.f ? 1.f / d : 0.f;
  float4 v = ((float4*)out)[i];
  v.x *= s; v.y *= s; v.z *= s; v.w *= s;
  ((float4*)out)[i] = v;
}

extern "C" void kernel_launch(void* const* d_in, const int* in_sizes, int n_in,
                              void* d_out, int out_size, void* d_ws, size_t ws_size,
                              hipStream_t stream) {
  const float*     x  = (const float*)d_in[0];
  const long long* ei = (const long long*)d_in[1];  // int64 edge_index [2, E]
  const float*     W  = (const float*)d_in[2];
  const float*     b  = (const float*)d_in[3];
  float* out = (float*)d_out;

  // Workspace layout: h [N_NODES*OUT_FEATS f32], then deg [N_NODES f32]
  float* h   = (float*)d_ws;
  float* deg = (float*)((char*)d_ws + (size_t)N_NODES * OUT_FEATS * sizeof(float));

  // 1) zero accumulators
  const int out4 = N_NODES * OUT_FEATS / 4;
  sage_zero_f4<<<(out4 + 255) / 256, 256, 0, stream>>>((float4*)out, out4);
  const int deg4 = N_NODES / 4;
  sage_zero_f4<<<(deg4 + 255) / 256, 256, 0, stream>>>((float4*)deg, deg4);

  // 2) h = x @ W + b  (WMMA)
  sage_gemm_wmma<<<N_NODES / 16, 512, 0, stream>>>(x, W, b, h);

  // 3) degree counts
  sage_degree<<<(N_EDGES + 255) / 256, 256, 0, stream>>>(ei, deg);

  // 4) edge scatter-add: 64 threads/edge
  const long long sthreads = (long long)N_EDGES * 64;
  sage_scatter<<<(unsigned)(sthreads / 256), 256, 0, stream>>>(h, ei, out);

  // 5) mean-normalize
  const int n4 = N_NODES * (OUT_FEATS / 4);
  sage_scale<<<(n4 + 255) / 256, 256, 0, stream>>>(out, deg);
}